// MambaBlock_34926674051721
// MI455X (gfx1250) — compile-verified
//
#include <hip/hip_runtime.h>
#include <hip/hip_bf16.h>
#include <math.h>

// ---------------- problem constants ----------------
#define DIM     1024
#define DSTATE  16
#define DCONV   4
#define BSZ     2
#define TLEN    4096
#define MTOT    (BSZ * TLEN)          // 8192 rows for all GEMMs
#define NCHUNK  32
#define TC      (TLEN / NCHUNK)       // 128 timesteps per chunk

typedef __attribute__((ext_vector_type(2))) float v2f;
typedef __attribute__((ext_vector_type(8))) float v8f;

// -------- gfx1250 async global->LDS copy (ASYNCcnt) with safe fallback --------
#if defined(__has_builtin)
#  if __has_builtin(__builtin_amdgcn_global_load_async_to_lds_b128) && \
      __has_builtin(__builtin_amdgcn_s_wait_asynccnt)
#    define USE_ASYNC_LDS 1
#  endif
#endif
#ifndef USE_ASYNC_LDS
#  define USE_ASYNC_LDS 0
#endif

#if USE_ASYNC_LDS
// Parameter type per hipcc diagnostic: 'vector_size(4*sizeof(int)) int __device__ *'
typedef int v4i_b128 __attribute__((vector_size(4 * sizeof(int))));
typedef __attribute__((address_space(1))) v4i_b128 glob_v4i;
typedef __attribute__((address_space(3))) v4i_b128 lds_v4i;
#endif

__device__ __forceinline__ void copy16_g2l(const float* gp, float* lp) {
#if USE_ASYNC_LDS
    __builtin_amdgcn_global_load_async_to_lds_b128(
        (glob_v4i*)gp, (lds_v4i*)lp, 0, 0);
#else
    *(float4*)lp = *(const float4*)gp;
#endif
}

__device__ __forceinline__ void wait_copies_and_barrier() {
#if USE_ASYNC_LDS
    __builtin_amdgcn_s_wait_asynccnt(0);
#endif
    __syncthreads();
}

// ---------------- tiled fp32 WMMA GEMM (LDS double-buffered) ----------------
// C[M,N] = A[M,K] * B[K,N], row-major fp32.  M%64==0, N%128==0, K%32==0.
// Block: 256 threads = 8 waves (4 M-waves x 2 N-waves); block tile 64x128;
// each wave computes a 16x64 subtile with V_WMMA_F32_16X16X4_F32.
// A-frag (16x4): lane l -> row (l&15); VGPR0 = col K+(l<16?0:2), VGPR1 = +1.
// B-frag (4x16): lane l -> col (l&15); VGPR0 = row K+(l<16?0:2), VGPR1 = +1.
// C/D: VGPR g, lane l -> row (l<16 ? g : 8+g), col (l&15).
#define BM 64
#define BN 128
#define BK 32
#define ASTRIDE 36   // padded A row stride in floats: 144B => 16B-aligned, bank-conflict-free
#define SLAB_FLOATS (BM * ASTRIDE + BK * BN)   // 2304 + 4096 = 6400

__device__ __forceinline__ void load_slab(const float* __restrict__ A,
                                          const float* __restrict__ Bm,
                                          int K, int N, int m0, int n0, int k0,
                                          float* sA, float* sB, int tid) {
    // A tile: 64x32 floats = 512 x b128 chunks, 2 per thread
#pragma unroll
    for (int r = 0; r < 2; ++r) {
        const int q = tid + r * 256;
        const int row = q >> 3;
        const int j = (q & 7) * 4;
        copy16_g2l(A + (size_t)(m0 + row) * K + k0 + j, sA + row * ASTRIDE + j);
    }
    // B tile: 32x128 floats = 1024 x b128 chunks, 4 per thread
#pragma unroll
    for (int r = 0; r < 4; ++r) {
        const int q = tid + r * 256;
        const int row = q >> 5;
        const int j = (q & 31) * 4;
        copy16_g2l(Bm + (size_t)(k0 + row) * N + n0 + j, sB + row * BN + j);
    }
}

__global__ __launch_bounds__(256)
void wmma_gemm_tiled(const float* __restrict__ A,
                     const float* __restrict__ Bm,
                     float* __restrict__ C,
                     int M, int N, int K) {
    __shared__ float smem[2 * SLAB_FLOATS];    // 51200 B of the 320KB WGP LDS

    const int tid  = threadIdx.x;
    const int lane = tid & 31;
    const int wave = tid >> 5;
    const int wm = wave & 3;          // 0..3 -> M subtile
    const int wn = wave >> 2;         // 0..1 -> N subtile

    const int nTiles = N / BN;
    const int mT = blockIdx.x / nTiles;
    const int nT = blockIdx.x % nTiles;
    const int m0 = mT * BM;
    const int n0 = nT * BN;

    const int khalf = (lane >> 4) * 2;   // 0 or 2
    const int mloc  = wm * 16 + (lane & 15);
    const int nloc  = wn * 64 + (lane & 15);

    v8f acc[4] = {};

    const int NS = K / BK;
    // prologue: slab 0 into buffer 0
    load_slab(A, Bm, K, N, m0, n0, 0, smem, smem + BM * ASTRIDE, tid);
    wait_copies_and_barrier();

    for (int s = 0; s < NS; ++s) {
        const int cur = s & 1;
        float* sA = smem + cur * SLAB_FLOATS;
        float* sB = sA + BM * ASTRIDE;
        if (s + 1 < NS) {
            float* nA = smem + (cur ^ 1) * SLAB_FLOATS;
            load_slab(A, Bm, K, N, m0, n0, (s + 1) * BK, nA, nA + BM * ASTRIDE, tid);
        }

        const float* aBase = sA + mloc * ASTRIDE + khalf;
        const float* bBase = sB + nloc;
#pragma unroll
        for (int kk = 0; kk < BK; kk += 4) {
            v2f a;
            a.x = aBase[kk];
            a.y = aBase[kk + 1];
#pragma unroll
            for (int t = 0; t < 4; ++t) {
                v2f b;
                b.x = bBase[(kk + khalf) * BN + t * 16];
                b.y = bBase[(kk + khalf + 1) * BN + t * 16];
                acc[t] = __builtin_amdgcn_wmma_f32_16x16x4_f32(
                    false, a, false, b, (short)0, acc[t], false, false);
            }
        }
        wait_copies_and_barrier();
    }

    const int rbase = m0 + wm * 16 + ((lane >> 4) ? 8 : 0);
    const int cbase = n0 + wn * 64 + (lane & 15);
#pragma unroll
    for (int t = 0; t < 4; ++t) {
#pragma unroll
        for (int g = 0; g < 8; ++g) {
            C[(size_t)(rbase + g) * N + cbase + t * 16] = acc[t][g];
        }
    }
}

// ---------------- skinny fp32 WMMA GEMM (direct loads), for N=32 ----------------
template <int NSUB>
__global__ __launch_bounds__(256)
void wmma_gemm_skinny(const float* __restrict__ A,
                      const float* __restrict__ Bm,
                      float* __restrict__ C,
                      int M, int N, int K) {
    const int wavesPerBlock = blockDim.x >> 5;
    const int wave = blockIdx.x * wavesPerBlock + (threadIdx.x >> 5);
    const int lane = threadIdx.x & 31;

    const int nTiles = N / (16 * NSUB);
    const int mTile  = wave / nTiles;
    const int nTile  = wave % nTiles;
    if (mTile * 16 >= M) return;

    const int m0 = mTile * 16;
    const int n0 = nTile * 16 * NSUB;

    v8f acc[NSUB] = {};
    const int mrow  = m0 + (lane & 15);
    const int khalf = (lane >> 4) * 2;
    const int ncol  = lane & 15;
    const float* __restrict__ arow = A + (size_t)mrow * K;

    for (int k = 0; k < K; k += 4) {
        v2f a;
        a.x = arow[k + khalf];
        a.y = arow[k + khalf + 1];
#pragma unroll
        for (int t = 0; t < NSUB; ++t) {
            v2f b;
            const size_t bb = (size_t)(k + khalf) * N + n0 + t * 16 + ncol;
            b.x = Bm[bb];
            b.y = Bm[bb + N];
            acc[t] = __builtin_amdgcn_wmma_f32_16x16x4_f32(
                false, a, false, b, (short)0, acc[t], false, false);
        }
    }
    const int rbase = m0 + ((lane >> 4) ? 8 : 0);
#pragma unroll
    for (int t = 0; t < NSUB; ++t)
#pragma unroll
        for (int g = 0; g < 8; ++g)
            C[(size_t)(rbase + g) * N + n0 + t * 16 + (lane & 15)] = acc[t][g];
}

// ---------------- depthwise causal conv + bias + SiLU ----------------
__global__ __launch_bounds__(256)
void conv_silu_kernel(const float* __restrict__ xz,
                      const float* __restrict__ conv_w,
                      const float* __restrict__ conv_b,
                      float* __restrict__ xc) {
    const int idx = blockIdx.x * blockDim.x + threadIdx.x;
    if (idx >= BSZ * TLEN * DIM) return;
    const int d = idx % DIM;
    const int t = (idx / DIM) % TLEN;
    const int b = idx / (DIM * TLEN);

    float acc = conv_b[d];
#pragma unroll
    for (int j = 0; j < DCONV; ++j) {
        const int tt = t - (DCONV - 1) + j;
        if (tt >= 0)
            acc += conv_w[d * DCONV + j] *
                   xz[((size_t)(b * TLEN + tt)) * (2 * DIM) + d];
    }
    xc[idx] = acc / (1.f + __expf(-acc));   // silu
}

__device__ __forceinline__ float softplus_f(float v) {
    return (v > 20.f) ? v : log1pf(__expf(v));
}

// ---------------- scan phase 1: per-chunk local scan ----------------
__global__ __launch_bounds__(256)
void scan_phase1(const float* __restrict__ dt_pre,
                 const float* __restrict__ b_dt,
                 const float* __restrict__ BC,
                 const float* __restrict__ xc,
                 const float* __restrict__ A_log,
                 float* __restrict__ P,
                 float* __restrict__ L) {
    const int c    = blockIdx.x % NCHUNK;
    const int dblk = (blockIdx.x / NCHUNK) % (DIM / 256);
    const int b    = blockIdx.x / (NCHUNK * (DIM / 256));
    const int d    = dblk * 256 + threadIdx.x;
    const int t0   = c * TC;

    float Arow[DSTATE];
#pragma unroll
    for (int n = 0; n < DSTATE; ++n)
        Arow[n] = -__expf(A_log[d * DSTATE + n]);

    __shared__ float sB[TC][DSTATE];
    for (int i = threadIdx.x; i < TC * DSTATE; i += 256) {
        const int tt = i / DSTATE, nn = i % DSTATE;
        sB[tt][nn] = BC[((size_t)(b * TLEN + t0 + tt)) * (2 * DSTATE) + nn];
    }
    __syncthreads();

    float h[DSTATE] = {};
    float pr[DSTATE];
#pragma unroll
    for (int n = 0; n < DSTATE; ++n) pr[n] = 1.f;

    const float bd = b_dt[d];
    for (int t = 0; t < TC; ++t) {
        const size_t off = ((size_t)(b * TLEN + t0 + t)) * DIM + d;
        const float dt = softplus_f(dt_pre[off] + bd);
        const float xv = xc[off];
        const float dtx = dt * xv;
#pragma unroll
        for (int n = 0; n < DSTATE; ++n) {
            const float a = __expf(dt * Arow[n]);
            pr[n] *= a;
            h[n] = a * h[n] + dtx * sB[t][n];
        }
    }

    const size_t base = (((size_t)c * BSZ + b) * DIM + d) * DSTATE;
#pragma unroll
    for (int n = 0; n < DSTATE; ++n) {
        P[base + n] = pr[n];
        L[base + n] = h[n];
    }
}

// ---------------- scan phase 2: cross-chunk recurrence ----------------
__global__ __launch_bounds__(256)
void scan_phase2(const float* __restrict__ P,
                 const float* __restrict__ L,
                 float* __restrict__ Hin) {
    const int idx = blockIdx.x * blockDim.x + threadIdx.x;
    if (idx >= BSZ * DIM * DSTATE) return;
    float H = 0.f;
    for (int c = 0; c < NCHUNK; ++c) {
        const size_t o = (size_t)c * BSZ * DIM * DSTATE + idx;
        Hin[o] = H;
        H = P[o] * H + L[o];
    }
}

// ---------------- scan phase 3: seeded re-scan + y*silu(z) epilogue ----------------
__global__ __launch_bounds__(256)
void scan_phase3(const float* __restrict__ dt_pre,
                 const float* __restrict__ b_dt,
                 const float* __restrict__ BC,
                 const float* xc,                 // aliases y_out (per-thread RAW-safe)
                 const float* __restrict__ A_log,
                 const float* __restrict__ D_param,
                 const float* __restrict__ Hin,
                 const float* __restrict__ xz,    // z = xz[..., DIM:2*DIM]
                 float* y_out) {
    const int c    = blockIdx.x % NCHUNK;
    const int dblk = (blockIdx.x / NCHUNK) % (DIM / 256);
    const int b    = blockIdx.x / (NCHUNK * (DIM / 256));
    const int d    = dblk * 256 + threadIdx.x;
    const int t0   = c * TC;

    float Arow[DSTATE];
#pragma unroll
    for (int n = 0; n < DSTATE; ++n)
        Arow[n] = -__expf(A_log[d * DSTATE + n]);

    __shared__ float sB[TC][DSTATE];
    __shared__ float sC[TC][DSTATE];
    for (int i = threadIdx.x; i < TC * 2 * DSTATE; i += 256) {
        const int tt = i >> 5, nn = i & 31;
        const float v = BC[((size_t)(b * TLEN + t0 + tt)) * (2 * DSTATE) + nn];
        if (nn < DSTATE) sB[tt][nn] = v;
        else             sC[tt][nn - DSTATE] = v;
    }
    __syncthreads();

    float h[DSTATE];
    const size_t hbase = (((size_t)c * BSZ + b) * DIM + d) * DSTATE;
#pragma unroll
    for (int n = 0; n < DSTATE; ++n) h[n] = Hin[hbase + n];

    const float bd = b_dt[d];
    const float Dp = D_param[d];
    for (int t = 0; t < TC; ++t) {
        const size_t off = ((size_t)(b * TLEN + t0 + t)) * DIM + d;
        const float dt = softplus_f(dt_pre[off] + bd);
        const float xv = xc[off];
        const float dtx = dt * xv;
        float yt = 0.f;
#pragma unroll
        for (int n = 0; n < DSTATE; ++n) {
            const float a = __expf(dt * Arow[n]);
            h[n] = a * h[n] + dtx * sB[t][n];
            yt += h[n] * sC[t][n];
        }
        yt += Dp * xv;
        const float zv = xz[((size_t)(b * TLEN + t0 + t)) * (2 * DIM) + DIM + d];
        y_out[off] = yt * (zv / (1.f + __expf(-zv)));
    }
}

// ---------------- launcher ----------------
extern "C" void kernel_launch(void* const* d_in, const int* in_sizes, int n_in,
                              void* d_out, int out_size, void* d_ws, size_t ws_size,
                              hipStream_t stream) {
    const float* x      = (const float*)d_in[0];
    const float* W_in   = (const float*)d_in[1];
    const float* conv_w = (const float*)d_in[2];
    const float* conv_b = (const float*)d_in[3];
    const float* A_log  = (const float*)d_in[4];
    const float* D_par  = (const float*)d_in[5];
    const float* W_x    = (const float*)d_in[6];
    const float* W_dt   = (const float*)d_in[7];
    const float* b_dt   = (const float*)d_in[8];
    const float* W_out  = (const float*)d_in[9];
    float* out = (float*)d_out;

    char* ws = (char*)d_ws;
    const size_t szXZ  = (size_t)MTOT * 2 * DIM * sizeof(float);              // 64 MB
    const size_t szXC  = (size_t)MTOT * DIM * sizeof(float);                  // 32 MB
    const size_t szBC  = (size_t)MTOT * 2 * DSTATE * sizeof(float);           //  1 MB
    const size_t szDT  = (size_t)MTOT * DIM * sizeof(float);                  // 32 MB
    const size_t szST  = (size_t)NCHUNK * BSZ * DIM * DSTATE * sizeof(float); //  4 MB

    float* xz     = (float*)(ws);
    float* xc     = (float*)(ws + szXZ);                       // reused as y_out
    float* BCbuf  = (float*)(ws + szXZ + szXC);
    float* dtbuf  = (float*)(ws + szXZ + szXC + szBC);
    float* Pbuf   = (float*)(ws + szXZ + szXC + szBC + szDT);
    float* Lbuf   = (float*)(ws + szXZ + szXC + szBC + szDT + szST);
    float* HinBuf = (float*)(ws + szXZ + szXC + szBC + szDT + 2 * szST);

    const int THREADS = 256;

    // 1) xz = x @ W_in   (8192 x 2048, K=1024)
    wmma_gemm_tiled<<<(MTOT / BM) * (2 * DIM / BN), THREADS, 0, stream>>>(
        x, W_in, xz, MTOT, 2 * DIM, DIM);

    // 2) depthwise conv + SiLU
    {
        const int n = BSZ * TLEN * DIM;
        conv_silu_kernel<<<(n + THREADS - 1) / THREADS, THREADS, 0, stream>>>(
            xz, conv_w, conv_b, xc);
    }

    // 3) BC = xc @ W_x   (8192 x 32, K=1024)
    {
        const int waves = (MTOT / 16) * (2 * DSTATE / 32);
        wmma_gemm_skinny<2><<<(waves + 7) / 8, THREADS, 0, stream>>>(
            xc, W_x, BCbuf, MTOT, 2 * DSTATE, DIM);
    }

    // 4) dt_pre = xc @ W_dt  (8192 x 1024, K=1024)
    wmma_gemm_tiled<<<(MTOT / BM) * (DIM / BN), THREADS, 0, stream>>>(
        xc, W_dt, dtbuf, MTOT, DIM, DIM);

    // 5) chunked scan, phase 1 (local scans)
    scan_phase1<<<BSZ * (DIM / 256) * NCHUNK, THREADS, 0, stream>>>(
        dtbuf, b_dt, BCbuf, xc, A_log, Pbuf, Lbuf);

    // 6) phase 2 (cross-chunk recurrence)
    {
        const int n = BSZ * DIM * DSTATE;
        scan_phase2<<<(n + THREADS - 1) / THREADS, THREADS, 0, stream>>>(
            Pbuf, Lbuf, HinBuf);
    }

    // 7) phase 3 (seeded re-scan + y*silu(z)), writes y over xc
    scan_phase3<<<BSZ * (DIM / 256) * NCHUNK, THREADS, 0, stream>>>(
        dtbuf, b_dt, BCbuf, xc, A_log, D_par, HinBuf, xz, xc);

    // 8) out = y @ W_out  (8192 x 1024, K=1024)
    wmma_gemm_tiled<<<(MTOT / BM) * (DIM / BN), THREADS, 0, stream>>>(
        xc, W_out, out, MTOT, DIM, DIM);

    (void)in_sizes; (void)n_in; (void)out_size; (void)ws_size;
}